// SilhouetteLoss_32366873543193
// MI455X (gfx1250) — compile-verified
//
#include <hip/hip_runtime.h>
#include <hip/hip_bf16.h>
#include <math.h>

typedef float v2f __attribute__((ext_vector_type(2)));
typedef float v8f __attribute__((ext_vector_type(8)));

namespace {
constexpr int   kH      = 240;            // 1920/8
constexpr int   kW      = 135;            // 1080/8
constexpr int   kNPix   = kH * kW;        // 32400
constexpr int   kNV     = 600;
constexpr int   kNF     = 1000;
constexpr int   kNFPad  = 1024;           // faces padded to multiple of 16
constexpr int   kNGroup = kNFPad / 16;    // 64 face groups of 16
constexpr float kSigmaInv = 1.0e4f;       // 1/SIGMA
constexpr float kBlur     = 9.21024036697585e-4f; // log(1/1e-4 - 1) * 1e-4

// workspace layout (float offsets)
constexpr int kWsVX   = 0;
constexpr int kWsVY   = 1024;
constexpr int kWsVZ   = 2048;
constexpr int kWsCoef = 3072;                         // 9 * 1024 * 4 = 36864 floats
constexpr int kWsEcon = kWsCoef + 9 * kNFPad * 4;     // 3 * 1024 * 2 = 6144 floats
} // namespace

// ---------------------------------------------------------------------------
// Kernel 1: project vertices, build per-face/per-edge polynomial coefficient
// columns for the WMMA B-matrices, plus |ab|^2 and 1/(|ab|^2+1e-12).
// Invalid faces (tz <= 1e-6) and padding faces get a coefficient pattern that
// yields inside=false and d2=1e30 -> contribution exactly 0.
// ---------------------------------------------------------------------------
__global__ __launch_bounds__(1024)
void sil_setup_kernel(const float* __restrict__ verts,
                      const int*   __restrict__ faces,
                      float*       __restrict__ ws,
                      float*       __restrict__ out) {
  const int t = threadIdx.x;
  if (t == 0) out[0] = 0.0f;   // loss accumulator

  float* vx = ws + kWsVX;
  float* vy = ws + kWsVY;
  float* vz = ws + kWsVZ;

  if (t < kNV) {
    // R = diag(-1,-1,1)
    float x = -verts[3 * t + 0];
    float y = -verts[3 * t + 1];
    float z =  verts[3 * t + 2];
    float zc = fmaxf(z, 1e-6f);
    vx[t] = (1000.0f * x / zc + 512.0f) * ((float)kW / 1024.0f);
    vy[t] = (1000.0f * y / zc + 512.0f) * ((float)kH / 1024.0f);
    vz[t] = z;                            // tz uses un-clamped z
  }
  __syncthreads();

  float* coef = ws + kWsCoef;
  float* econ = ws + kWsEcon;

  const int f = t;
  if (f < kNFPad) {
    float ax[3], ay[3];
    bool ok = false;
    if (f < kNF) {
      int i0 = faces[3 * f + 0], i1 = faces[3 * f + 1], i2 = faces[3 * f + 2];
      ax[0] = vx[i0]; ay[0] = vy[i0];
      ax[1] = vx[i1]; ay[1] = vy[i1];
      ax[2] = vx[i2]; ay[2] = vy[i2];
      float tz = (vz[i0] + vz[i1] + vz[i2]) * (1.0f / 3.0f);
      ok = (tz > 1e-6f);
    }
    for (int k = 0; k < 3; ++k) {
      float c[3][4];
      float ab2, inv;
      if (ok) {
        float aX = ax[k], aY = ay[k];
        float bX = ax[(k + 1) % 3], bY = ay[(k + 1) % 3];
        float abx = bX - aX, aby = bY - aY;
        // cross = ab.x*(py-a.y) - ab.y*(px-a.x) = c0 + c1*px + c2*py
        c[0][0] = aby * aX - abx * aY; c[0][1] = -aby;        c[0][2] = abx;        c[0][3] = 0.0f;
        // ap.ab = d0 + d1*px + d2*py
        c[1][0] = -(abx * aX + aby * aY); c[1][1] = abx;      c[1][2] = aby;        c[1][3] = 0.0f;
        // |ap|^2 = a0 + a1*px + a2*py + 1*(px^2+py^2)
        c[2][0] = aX * aX + aY * aY; c[2][1] = -2.0f * aX;    c[2][2] = -2.0f * aY; c[2][3] = 1.0f;
        ab2 = abx * abx + aby * aby;
        inv = 1.0f / (ab2 + 1e-12f);
      } else {
        // force pos=false (edge1 cross=-1), neg=false (edge0 cross=+1),
        // d2 = 1e30 > BLUR  -> prob = 0
        for (int j = 0; j < 3; ++j)
          for (int r = 0; r < 4; ++r) c[j][r] = 0.0f;
        c[0][0] = (k == 0) ? 1.0f : ((k == 1) ? -1.0f : 0.0f);
        c[2][0] = 1.0e30f;
        ab2 = 1.0f; inv = 1.0f;
      }
      for (int j = 0; j < 3; ++j) {
        int q = k * 3 + j;
        float* dst = coef + (((q * kNFPad) + f) << 2);
        dst[0] = c[j][0]; dst[1] = c[j][1]; dst[2] = c[j][2]; dst[3] = c[j][3];
      }
      float* ed = econ + (((k * kNFPad) + f) << 1);
      ed[0] = ab2; ed[1] = inv;
    }
  }
}

// ---------------------------------------------------------------------------
// Kernel 2: one wave32 per 16-pixel tile. For each group of 16 faces, nine
// V_WMMA_F32_16X16X4_F32 ops evaluate {cross, ap.ab, |ap|^2} for all 3 edges
// over the 16x16 pixel-face tile; fp32 VALU epilogue does clamp/d2/softplus
// with native v_exp_f32/v_log_f32 transcendentals.
// A fragment (16x4 f32, ISA layout): lanes 0-15 hold K0,K1; lanes 16-31 K2,K3.
// C/D (16x16 f32): vgpr i -> row M = i + (lane>=16 ? 8 : 0), col N = lane%16.
// ---------------------------------------------------------------------------
__global__ __launch_bounds__(256)
void sil_raster_kernel(const float* __restrict__ ws,
                       const float* __restrict__ gt,
                       float*       __restrict__ out) {
  const float* coef = ws + kWsCoef;
  const float* econ = ws + kWsEcon;

  const int  lane = threadIdx.x & 31;
  const int  wave = threadIdx.x >> 5;
  const int  tile = blockIdx.x * 8 + wave;
  const int  m    = lane & 15;          // pixel row (A) / face col (B,C)
  const bool hi   = lane >= 16;

  // pixel basis row A[m] = [1, px, py, px^2+py^2]
  int pid = tile * 16 + m;
  if (pid > kNPix - 1) pid = kNPix - 1;     // clamp; writes guarded later
  float px = (float)(pid % kW) + 0.5f;
  float py = (float)(pid / kW) + 0.5f;
  v2f Af;
  if (!hi) { Af.x = 1.0f; Af.y = px; }
  else     { Af.x = py;   Af.y = px * px + py * py; }

  float S[8];
  #pragma unroll
  for (int i = 0; i < 8; ++i) S[i] = 0.0f;

  for (int g = 0; g < kNGroup; ++g) {
    const int f   = g * 16 + m;          // face for this lane's B column
    const int off = hi ? 2 : 0;          // lanes 0-15: (K0,K1); 16-31: (K2,K3)

    // keep next group's coefficient lines warm (global_prefetch_b8)
    __builtin_prefetch(coef + (((0 * kNFPad) + f + 16) << 2), 0, 1);

    v8f acc[9];
    #pragma unroll
    for (int q = 0; q < 9; ++q) {        // q = edge*3 + {cross,dot,ap2}
      v2f Bf = *(const v2f*)(coef + (((q * kNFPad) + f) << 2) + off);
      v8f cz = {};
      acc[q] = __builtin_amdgcn_wmma_f32_16x16x4_f32(
          false, Af, false, Bf, (short)0, cz, false, false);
    }

    v2f e0 = *(const v2f*)(econ + (((0 * kNFPad) + f) << 1));  // (|ab|^2, inv)
    v2f e1 = *(const v2f*)(econ + (((1 * kNFPad) + f) << 1));
    v2f e2 = *(const v2f*)(econ + (((2 * kNFPad) + f) << 1));

    #pragma unroll
    for (int i = 0; i < 8; ++i) {
      float cr0 = acc[0][i], dt0 = acc[1][i], a20 = acc[2][i];
      float cr1 = acc[3][i], dt1 = acc[4][i], a21 = acc[5][i];
      float cr2 = acc[6][i], dt2 = acc[7][i], a22 = acc[8][i];

      // inside = all crosses >= 0 or all crosses <= 0  (min3/max3 form)
      float crmn = fminf(fminf(cr0, cr1), cr2);
      float crmx = fmaxf(fmaxf(cr0, cr1), cr2);
      bool inside = (crmn >= 0.0f) || (crmx <= 0.0f);

      float t0 = fminf(fmaxf(dt0 * e0.y, 0.0f), 1.0f);
      float t1 = fminf(fmaxf(dt1 * e1.y, 0.0f), 1.0f);
      float t2 = fminf(fmaxf(dt2 * e2.y, 0.0f), 1.0f);
      // d2 = |ap|^2 + t*(t*|ab|^2 - 2*ap.ab)
      float d20 = fmaf(t0, fmaf(t0, e0.x, -2.0f * dt0), a20);
      float d21 = fmaf(t1, fmaf(t1, e1.x, -2.0f * dt1), a21);
      float d22 = fmaf(t2, fmaf(t2, e2.x, -2.0f * dt2), a22);
      float d2min = fminf(fminf(d20, d21), d22);

      bool valid = inside || (d2min <= kBlur);
      // u = -signed_d2/sigma ; log1p(-sigmoid(u)) = -softplus(u)
      float u  = (inside ? d2min : -d2min) * kSigmaInv;
      float e  = __expf(-fabsf(u));                  // v_exp_f32
      float sp = fmaxf(u, 0.0f) + __logf(1.0f + e);  // v_log_f32
      S[i] -= valid ? sp : 0.0f;
    }
  }

  // reduce log-sums over the 16 faces held across each 16-lane half
  #pragma unroll
  for (int mask = 1; mask < 16; mask <<= 1) {
    #pragma unroll
    for (int i = 0; i < 8; ++i) S[i] += __shfl_xor(S[i], mask, 32);
  }

  if (m == 0) {  // lanes 0 and 16 each own 8 pixel rows
    float lsum = 0.0f;
    #pragma unroll
    for (int i = 0; i < 8; ++i) {
      int pixel = tile * 16 + i + (hi ? 8 : 0);
      if (pixel < kNPix) {
        float alpha = 1.0f - __expf(S[i]);
        out[1 + pixel] = alpha;
        lsum += fabsf(alpha - gt[pixel]);
      }
    }
    atomicAdd(out, lsum * (1.0f / (float)kNPix));
  }
}

extern "C" void kernel_launch(void* const* d_in, const int* in_sizes, int n_in,
                              void* d_out, int out_size, void* d_ws, size_t ws_size,
                              hipStream_t stream) {
  (void)in_sizes; (void)n_in; (void)out_size; (void)ws_size;
  const float* verts = (const float*)d_in[0];   // (600,3) f32
  const float* gt    = (const float*)d_in[1];   // (240,135) f32
  const int*   faces = (const int*)d_in[2];     // (1000,3) i32
  float* out = (float*)d_out;                   // [loss, sil(32400)]
  float* ws  = (float*)d_ws;                    // ~184 KB used

  sil_setup_kernel<<<dim3(1), dim3(1024), 0, stream>>>(verts, faces, ws, out);

  const int ntiles  = (kNPix + 15) / 16;        // 2025 tiles of 16 pixels
  const int nblocks = (ntiles + 7) / 8;         // 8 waves per 256-thread block
  sil_raster_kernel<<<dim3(nblocks), dim3(256), 0, stream>>>(ws, gt, out);
}